// BahdanauAttention_33002528702914
// MI455X (gfx1250) — compile-verified
//
#include <hip/hip_runtime.h>

typedef __attribute__((ext_vector_type(2))) float v2f;
typedef __attribute__((ext_vector_type(8))) float v8f;

#define BB 8
#define TT 64
#define SS 512
#define QU 1024
#define VU 512
#define UU 256

// ---------------------------------------------------------------------------
// Generic (optionally batched) f32 GEMM on V_WMMA_F32_16X16X4_F32.
//   C[b] (M,N) = A[b] (M,K) * B[b] (K,N),  row-major, lda=K, ldb=N, ldc=N.
// One wave computes one 16x16 C tile; 8 waves (256 threads) per block.
// ---------------------------------------------------------------------------
__global__ __launch_bounds__(256) void wmma_gemm_f32(
    const float* __restrict__ A, const float* __restrict__ Bm,
    float* __restrict__ C, int M, int N, int K,
    long strideA, long strideB, long strideC)
{
    const int batch = blockIdx.y;
    const float* Ab = A  + (long)batch * strideA;
    const float* Bb = Bm + (long)batch * strideB;
    float*       Cb = C  + (long)batch * strideC;

    const int lane   = threadIdx.x & 31;
    const int wave   = threadIdx.x >> 5;
    const int tilesN = N >> 4;
    const int tilesM = M >> 4;
    const int tile   = blockIdx.x * 8 + wave;       // wave-uniform
    if (tile >= tilesM * tilesN) return;            // uniform branch: EXEC stays full
    const int tm = (tile / tilesN) << 4;
    const int tn = (tile % tilesN) << 4;

    const int half = lane >> 4;                     // 0: K0..1 / 1: K2..3
    const int l16  = lane & 15;

    v8f acc = {};
    const float* arow = Ab + (long)(tm + l16) * K;  // this lane's A row
    const float* bcol = Bb + (tn + l16);            // this lane's B column

    for (int kk = 0; kk < K; kk += 4) {
        const int ka = kk + 2 * half;               // even -> 8B aligned
        v2f a = *(const v2f*)(arow + ka);
        v2f b;
        b.x = bcol[(long)ka       * N];
        b.y = bcol[(long)(ka + 1) * N];
        acc = __builtin_amdgcn_wmma_f32_16x16x4_f32(
            /*neg_a=*/false, a, /*neg_b=*/false, b,
            /*c_mod=*/(short)0, acc, /*reuse_a=*/false, /*reuse_b=*/false);
    }

    float* crow = Cb + (long)(tm + half * 8) * N + tn + l16;
#pragma unroll
    for (int v = 0; v < 8; ++v) crow[(long)v * N] = acc[v];
}

// ---------------------------------------------------------------------------
// Fused additive-attention scores + softmax.
// One workgroup per (b,t): scores[s] = sum_u scale[u]*tanh(w1q[bt,u]+w2k[bs,u])
// Each wave owns one s at a time (lane covers 8 u's via two float4 loads),
// wave32 shuffle reduction, then block softmax over the 512 scores in LDS.
// Normalized weights written straight to the attention half of d_out.
// ---------------------------------------------------------------------------
__global__ __launch_bounds__(256) void scores_softmax(
    const float* __restrict__ w1q,   // (B*T, UU)
    const float* __restrict__ w2k,   // (B*S, UU)
    const float* __restrict__ scale, // (UU)
    float* __restrict__ attn)        // (B*T, S)
{
    __shared__ float s_scores[SS];
    __shared__ float red[256];

    const int bt   = blockIdx.x;       // 0..B*T-1
    const int b    = bt >> 6;          // T = 64
    const int tid  = threadIdx.x;
    const int lane = tid & 31;
    const int wave = tid >> 5;

    // Per-lane slice of the query projection and scale vector (u = lane*8..+7)
    const float4* qrow = (const float4*)(w1q + (long)bt * UU) + lane * 2;
    const float4  q0 = qrow[0], q1 = qrow[1];
    const float4* srow = (const float4*)scale + lane * 2;
    const float4  c0 = srow[0], c1 = srow[1];

    for (int s = wave; s < SS; s += 8) {
        const float4* krow = (const float4*)(w2k + (long)(b * SS + s) * UU) + lane * 2;
        const float4 k0 = krow[0], k1 = krow[1];
        float p;
        p  = c0.x * tanhf(q0.x + k0.x);
        p += c0.y * tanhf(q0.y + k0.y);
        p += c0.z * tanhf(q0.z + k0.z);
        p += c0.w * tanhf(q0.w + k0.w);
        p += c1.x * tanhf(q1.x + k1.x);
        p += c1.y * tanhf(q1.y + k1.y);
        p += c1.z * tanhf(q1.z + k1.z);
        p += c1.w * tanhf(q1.w + k1.w);
#pragma unroll
        for (int off = 16; off; off >>= 1) p += __shfl_xor(p, off, 32);
        if (lane == 0) s_scores[s] = p;
    }
    __syncthreads();

    // Softmax over 512 scores: each thread owns 2.
    const float v0 = s_scores[tid];
    const float v1 = s_scores[tid + 256];
    red[tid] = fmaxf(v0, v1);
    __syncthreads();
#pragma unroll
    for (int off = 128; off; off >>= 1) {
        if (tid < off) red[tid] = fmaxf(red[tid], red[tid + off]);
        __syncthreads();
    }
    const float m = red[0];
    __syncthreads();

    const float e0 = __expf(v0 - m);
    const float e1 = __expf(v1 - m);
    red[tid] = e0 + e1;
    __syncthreads();
#pragma unroll
    for (int off = 128; off; off >>= 1) {
        if (tid < off) red[tid] += red[tid + off];
        __syncthreads();
    }
    const float inv = 1.0f / red[0];

    attn[(long)bt * SS + tid]       = e0 * inv;
    attn[(long)bt * SS + tid + 256] = e1 * inv;
}

// ---------------------------------------------------------------------------
// Launch: proj_q -> proj_k -> scores+softmax -> batched context GEMM
// ---------------------------------------------------------------------------
extern "C" void kernel_launch(void* const* d_in, const int* in_sizes, int n_in,
                              void* d_out, int out_size, void* d_ws, size_t ws_size,
                              hipStream_t stream)
{
    (void)in_sizes; (void)n_in; (void)out_size; (void)ws_size;

    const float* query = (const float*)d_in[0]; // (B,T,QU)
    const float* value = (const float*)d_in[1]; // (B,S,VU)
    const float* W1    = (const float*)d_in[2]; // (QU,UU)
    const float* W2    = (const float*)d_in[3]; // (VU,UU)
    const float* scale = (const float*)d_in[4]; // (UU)

    float* ctx  = (float*)d_out;                 // (B,T,VU)
    float* attn = ctx + (long)BB * TT * VU;      // (B,T,S)

    float* w1q = (float*)d_ws;                   // (B*T, UU) = 512 KB
    float* w2k = w1q + (long)BB * TT * UU;       // (B*S, UU) = 4 MB

    // w1q = query @ W1 : M=512, N=256, K=1024 -> 512 tiles / 8 waves = 64 blocks
    wmma_gemm_f32<<<dim3(64, 1), 256, 0, stream>>>(
        query, W1, w1q, BB * TT, UU, QU, 0, 0, 0);

    // w2k = value @ W2 : M=4096, N=256, K=512 -> 4096 tiles -> 512 blocks
    wmma_gemm_f32<<<dim3(512, 1), 256, 0, stream>>>(
        value, W2, w2k, BB * SS, UU, VU, 0, 0, 0);

    // scores + softmax: one WG per (b,t)
    scores_softmax<<<dim3(BB * TT), 256, 0, stream>>>(w1q, w2k, scale, attn);

    // ctx[b] = attn[b] (64x512) @ value[b] (512x512) : 128 tiles -> 16 blocks, batch=8
    wmma_gemm_f32<<<dim3(16, BB), 256, 0, stream>>>(
        attn, value, ctx, TT, VU, SS,
        (long)TT * SS, (long)SS * VU, (long)TT * VU);
}